// PredictiveMemoryBank_39582418600443
// MI455X (gfx1250) — compile-verified
//
#include <hip/hip_runtime.h>
#include <math.h>

// ---------------------------------------------------------------------------
// Problem constants (match reference setup_inputs)
// ---------------------------------------------------------------------------
#define BATCH   2
#define SEQ     8192
#define HDIM    2048
#define PDIM    512
#define NSLOTS  64
#define ROWS    (BATCH * SEQ)          // 16384 rows to reduce
#define NCHUNK  64                     // row chunks for pass-1 reduction
#define RPC     (ROWS / NCHUNK)        // 256 rows per chunk

typedef __attribute__((ext_vector_type(2))) float v2f;
typedef __attribute__((ext_vector_type(8))) float v8f;

// ---------------------------------------------------------------------------
// Kernel 1: column-sum of hidden_states via V_WMMA_F32_16X16X4_F32.
// D = A(16x4 ones) * B(4x16 tile of hidden) + C  => every row of D holds the
// 16 column sums. One wave (32 lanes) per (16-col tile, row chunk).
// B f32 layout: lane l holds column (l&15); lanes 0-15 carry K={0,1},
// lanes 16-31 carry K={2,3} (any K permutation is fine: A is all-ones).
// ---------------------------------------------------------------------------
__global__ __launch_bounds__(32)
void colsum_wmma_kernel(const float* __restrict__ X, float* __restrict__ partials) {
    const int lane  = threadIdx.x;          // 0..31, full wave active (WMMA req.)
    const int col0  = blockIdx.x * 16;      // 128 column tiles
    const int row0  = blockIdx.y * RPC;     // 64 row chunks
    const int n     = lane & 15;
    const int khalf = lane >> 4;            // 0 -> rows {r,r+1}, 1 -> rows {r+2,r+3}

    v2f a; a.x = 1.0f; a.y = 1.0f;          // ones fragment (layout-invariant)
    v8f c = {};                             // zero accumulator

    const float* base = X + (size_t)(row0 + 2 * khalf) * HDIM + col0 + n;
    #pragma unroll 4
    for (int r = 0; r < RPC; r += 4) {
        v2f b;
        b.x = base[0];                      // row r + 2*khalf
        b.y = base[HDIM];                   // row r + 2*khalf + 1
        c = __builtin_amdgcn_wmma_f32_16x16x4_f32(
                /*neg_a=*/false, a, /*neg_b=*/false, b,
                /*c_mod=*/(short)0, c, /*reuse_a=*/false, /*reuse_b=*/false);
        base += (size_t)4 * HDIM;
    }
    // D VGPR0, lanes 0-15 = D[0][n] = column sums (all D rows identical).
    if (lane < 16)
        partials[(size_t)blockIdx.y * HDIM + col0 + lane] = c[0];
}

// ---------------------------------------------------------------------------
// Kernel 2: deterministic second-pass reduction -> actual[h] (the mean).
// ---------------------------------------------------------------------------
__global__ __launch_bounds__(256)
void finalize_actual_kernel(const float* __restrict__ partials, float* __restrict__ actual) {
    int h = blockIdx.x * blockDim.x + threadIdx.x;
    if (h >= HDIM) return;
    float s = 0.0f;
    #pragma unroll 8
    for (int i = 0; i < NCHUNK; ++i) s += partials[(size_t)i * HDIM + h];
    actual[h] = s * (1.0f / (float)ROWS);
}

// ---------------------------------------------------------------------------
// Generic GEMV: y[r] = dot(W[r,:], x) (+ bias). One wave per row,
// lane-strided float4 loads (coalesced 512B per wave), wave32 shfl reduce.
// C must be a multiple of 128 (512/2048 both are).
// ---------------------------------------------------------------------------
__global__ __launch_bounds__(256)
void gemv_kernel(const float* __restrict__ W, const float* __restrict__ x,
                 const float* __restrict__ bias, float* __restrict__ y,
                 int R, int C) {
    int gwave = (blockIdx.x * blockDim.x + threadIdx.x) >> 5;
    int lane  = threadIdx.x & 31;
    if (gwave >= R) return;
    const float* row = W + (size_t)gwave * C;
    float s = 0.0f;
    for (int j = lane * 4; j < C; j += 128) {
        float4 w = *(const float4*)(row + j);
        float4 v = *(const float4*)(x + j);
        s += w.x * v.x + w.y * v.y + w.z * v.z + w.w * v.w;
    }
    #pragma unroll
    for (int off = 16; off > 0; off >>= 1) s += __shfl_xor(s, off, 32);
    if (lane == 0) y[gwave] = s + (bias ? bias[gwave] : 0.0f);
}

// ---------------------------------------------------------------------------
// Kernel: prediction error, norm, surprise (single block, deterministic).
// ---------------------------------------------------------------------------
__global__ __launch_bounds__(256)
void err_surprise_kernel(const float* __restrict__ actual,
                         const float* __restrict__ predicted,
                         const float* __restrict__ scale,
                         float* __restrict__ out_surprise,
                         float* __restrict__ out_err,
                         float* __restrict__ ws_surprise) {
    __shared__ float red[8];
    float local = 0.0f;
    for (int h = threadIdx.x; h < HDIM; h += 256) {
        float e = actual[h] - predicted[h];
        out_err[h] = e;
        local += e * e;
    }
    #pragma unroll
    for (int off = 16; off > 0; off >>= 1) local += __shfl_xor(local, off, 32);
    int wave = threadIdx.x >> 5, lane = threadIdx.x & 31;
    if (lane == 0) red[wave] = local;
    __syncthreads();
    if (threadIdx.x == 0) {
        float s = 0.0f;
        #pragma unroll
        for (int i = 0; i < 8; ++i) s += red[i];
        float norm = sqrtf(s * (1.0f / (float)HDIM));       // ||e|| / sqrt(H)
        float sp = 1.0f / (1.0f + expf(-scale[0] * norm));
        out_surprise[0] = sp;
        ws_surprise[0]  = sp;
    }
}

// ---------------------------------------------------------------------------
// Kernel: GRU gate math (PyTorch order r,z,n) -> new_pred_state.
// ---------------------------------------------------------------------------
__global__ __launch_bounds__(256)
void gates_kernel(const float* __restrict__ gi, const float* __restrict__ gh,
                  const float* __restrict__ hprev, float* __restrict__ out_h) {
    int p = blockIdx.x * blockDim.x + threadIdx.x;
    if (p >= PDIM) return;
    float r = 1.0f / (1.0f + expf(-(gi[p]            + gh[p])));
    float z = 1.0f / (1.0f + expf(-(gi[PDIM + p]     + gh[PDIM + p])));
    float n = tanhf(gi[2 * PDIM + p] + r * gh[2 * PDIM + p]);
    out_h[p] = (1.0f - z) * n + z * hprev[p];
}

// ---------------------------------------------------------------------------
// Kernel: memory bank decay + one-hot slot write.
// ---------------------------------------------------------------------------
__global__ __launch_bounds__(256)
void mem_kernel(const float* __restrict__ mem_states,
                const float* __restrict__ mem_strength,
                const int* __restrict__ wptr,
                const float* __restrict__ actual,
                const float* __restrict__ sp_ws,
                float* __restrict__ out_states,
                float* __restrict__ out_strength,
                float decay_pow) {
    int idx = blockIdx.x * blockDim.x + threadIdx.x;
    int slot = ((wptr[0] % NSLOTS) + NSLOTS) % NSLOTS;
    float sp = sp_ws[0];
    if (idx < NSLOTS * HDIM) {
        int row = idx >> 11;                // /HDIM
        int col = idx & (HDIM - 1);
        out_states[idx] = (row == slot) ? actual[col] * sp : mem_states[idx];
    }
    if (idx < NSLOTS)
        out_strength[idx] = (idx == slot) ? sp : mem_strength[idx] * decay_pow;
}

// ---------------------------------------------------------------------------
// Launch
// ---------------------------------------------------------------------------
extern "C" void kernel_launch(void* const* d_in, const int* in_sizes, int n_in,
                              void* d_out, int out_size, void* d_ws, size_t ws_size,
                              hipStream_t stream) {
    const float* hidden       = (const float*)d_in[0];
    const float* pred_state   = (const float*)d_in[1];
    const float* mem_states   = (const float*)d_in[2];
    const float* mem_strength = (const float*)d_in[3];
    const float* W_ih         = (const float*)d_in[4];
    const float* W_hh         = (const float*)d_in[5];
    const float* b_ih         = (const float*)d_in[6];
    const float* b_hh         = (const float*)d_in[7];
    const float* pred_proj_w  = (const float*)d_in[8];
    const float* input_proj_w = (const float*)d_in[9];
    const float* s_scale      = (const float*)d_in[10];
    const int*   write_ptr    = (const int*)d_in[11];

    float* out = (float*)d_out;
    // Output layout (flat, reference return order):
    float* o_surprise = out;                               // 1
    float* o_err      = out + 1;                           // 2048
    float* o_pred     = out + 1 + HDIM;                    // 2048
    float* o_mem      = out + 1 + 2 * HDIM;                // 64*2048
    float* o_str      = o_mem + NSLOTS * HDIM;             // 64
    float* o_h        = o_str + NSLOTS;                    // 512

    // Workspace layout (floats)
    float* ws = (float*)d_ws;
    float* w_partials = ws;                                // NCHUNK*HDIM = 131072
    float* w_actual   = w_partials + NCHUNK * HDIM;        // 2048
    float* w_xgru     = w_actual + HDIM;                   // 2048
    float* w_gi       = w_xgru + HDIM;                     // 1536
    float* w_gh       = w_gi + 3 * PDIM;                   // 1536
    float* w_surprise = w_gh + 3 * PDIM;                   // 1

    const float decay_pow = (float)pow(0.999, (double)SEQ);

    // 1) hidden mean: WMMA partial column sums, then deterministic finalize
    colsum_wmma_kernel<<<dim3(HDIM / 16, NCHUNK), 32, 0, stream>>>(hidden, w_partials);
    finalize_actual_kernel<<<HDIM / 256, 256, 0, stream>>>(w_partials, w_actual);

    // 2) predicted = pred_proj_w @ pred_state   (R=2048, C=512)
    gemv_kernel<<<HDIM / 8, 256, 0, stream>>>(pred_proj_w, pred_state, nullptr,
                                              o_pred, HDIM, PDIM);

    // 3) error, norm, surprise
    err_surprise_kernel<<<1, 256, 0, stream>>>(w_actual, o_pred, s_scale,
                                               o_surprise, o_err, w_surprise);

    // 4) x_gru = input_proj_w @ actual          (R=2048, C=2048)
    gemv_kernel<<<HDIM / 8, 256, 0, stream>>>(input_proj_w, w_actual, nullptr,
                                              w_xgru, HDIM, HDIM);

    // 5) gi = W_ih @ x_gru + b_ih               (R=1536, C=2048)
    gemv_kernel<<<(3 * PDIM) / 8, 256, 0, stream>>>(W_ih, w_xgru, b_ih,
                                                    w_gi, 3 * PDIM, HDIM);

    // 6) gh = W_hh @ pred_state + b_hh          (R=1536, C=512)
    gemv_kernel<<<(3 * PDIM) / 8, 256, 0, stream>>>(W_hh, pred_state, b_hh,
                                                    w_gh, 3 * PDIM, PDIM);

    // 7) GRU gates -> new_pred_state
    gates_kernel<<<(PDIM + 255) / 256, 256, 0, stream>>>(w_gi, w_gh, pred_state, o_h);

    // 8) memory bank update
    mem_kernel<<<(NSLOTS * HDIM + 255) / 256, 256, 0, stream>>>(
        mem_states, mem_strength, write_ptr, w_actual, w_surprise,
        o_mem, o_str, decay_pow);
}